// ImageWarpingLayer_53309134078368
// MI455X (gfx1250) — compile-verified
//
#include <hip/hip_runtime.h>

// Dense image warp (bilinear gather), NHWC fp32.
// B=16, H=512, W=512, C=8 fixed by the reference; H*W = 2^18 lets us use
// shift/mask index decomposition on wave32. All buffers < 4GB, so every
// address is computed as a 32-bit byte offset off a 64-bit SGPR base
// (saddr + voffset addressing; no 64-bit VALU address math).

typedef float v2f __attribute__((ext_vector_type(2)));
typedef float v4f __attribute__((ext_vector_type(4)));

__global__ __launch_bounds__(256) void warp_bilinear_kernel(
    const float* __restrict__ image,   // [B,512,512,8]
    const float* __restrict__ flow,    // [B,512,512,2]
    float* __restrict__ out,           // [B,512,512,8]
    int n_pix) {
  constexpr int H = 512, W = 512, C = 8;

  const unsigned pix = blockIdx.x * 256u + threadIdx.x;   // flat (b,y,x)
  if (pix >= (unsigned)n_pix) return;

  const unsigned x = pix & (W - 1);
  const unsigned y = (pix >> 9) & (H - 1);
  const unsigned b = pix >> 18;                           // H*W == 2^18

  // Flow is streamed once -> non-temporal load (keep L2 free for the image).
  const v2f f = __builtin_nontemporal_load(
      (const v2f*)((const char*)flow + (size_t)(pix * 8u)));

  const float qy = (float)y + f.x;                  // flow[...,0] = dy
  const float qx = (float)x + f.y;                  // flow[...,1] = dx

  // floor clamped to [0, size-2]; alpha = clip(q - floor, 0, 1)  (TF impl)
  float fy = floorf(qy); fy = fminf(fmaxf(fy, 0.0f), (float)(H - 2));
  float fx = floorf(qx); fx = fminf(fmaxf(fx, 0.0f), (float)(W - 2));
  const float ay = fminf(fmaxf(qy - fy, 0.0f), 1.0f);
  const float ax = fminf(fmaxf(qx - fx, 0.0f), 1.0f);
  const unsigned iy = (unsigned)(int)fy;
  const unsigned ix = (unsigned)(int)fx;

  // 32-bit byte offset of image[b, iy, ix, 0]; whole image is 134MB < 4GB.
  // Row of taps (tl|tr) is 64 contiguous bytes -> 4x global_load_b128.
  const unsigned byteOff = (((b << 9) + iy) * (unsigned)W + ix) * (C * 4u);
  const char* basep = (const char*)image + (size_t)byteOff;
  const v4f* toprow = (const v4f*)basep;                       // tl0 tl1 tr0 tr1
  const v4f* botrow = (const v4f*)(basep + (unsigned)(W * C * 4));  // bl0 bl1 br0 br1

  const v4f tl0 = toprow[0], tl1 = toprow[1];
  const v4f tr0 = toprow[2], tr1 = toprow[3];
  const v4f bl0 = botrow[0], bl1 = botrow[1];
  const v4f br0 = botrow[2], br1 = botrow[3];

  // top = tl + ax*(tr-tl); bot = bl + ax*(br-bl); out = top + ay*(bot-top)
  // Scalar * ext_vector splats; HIP contracts mul+add into v_fma_f32.
  const v4f top0 = tl0 + ax * (tr0 - tl0);
  const v4f top1 = tl1 + ax * (tr1 - tl1);
  const v4f bot0 = bl0 + ax * (br0 - bl0);
  const v4f bot1 = bl1 + ax * (br1 - bl1);
  const v4f o0 = top0 + ay * (bot0 - top0);
  const v4f o1 = top1 + ay * (bot1 - top1);

  // Output is write-once streaming -> non-temporal b128 stores, 32-bit offset.
  v4f* dst = (v4f*)((char*)out + (size_t)(pix * 32u));
  __builtin_nontemporal_store(o0, dst);
  __builtin_nontemporal_store(o1, dst + 1);
}

extern "C" void kernel_launch(void* const* d_in, const int* in_sizes, int n_in,
                              void* d_out, int out_size, void* d_ws, size_t ws_size,
                              hipStream_t stream) {
  const float* image = (const float*)d_in[0];  // [B,512,512,8] fp32
  const float* flow  = (const float*)d_in[1];  // [B,512,512,2] fp32
  float* out = (float*)d_out;                  // [B,512,512,8] fp32

  const int n_pix = in_sizes[1] / 2;           // B*H*W
  const int blocks = (n_pix + 255) / 256;

  warp_bilinear_kernel<<<blocks, 256, 0, stream>>>(image, flow, out, n_pix);
}